// S_TauNet_AVE_loghist_15298673509111
// MI455X (gfx1250) — compile-verified
//
#include <hip/hip_runtime.h>
#include <cstdint>

#define EPSV 1e-5f
#define BS 64
#define XPITCH 84   // 80 + 4 pad: bank stride 84 -> only 2-way LDS conflicts, 16B-aligned rows

// ---------------- CDNA5 async global->LDS helpers ----------------
// Builtin prototypes (from hipcc diagnostics, round 1):
//   b128: arg0 = int __vector_size(16) addrspace(1)*  (global src)
//   b32 : arg0 = int addrspace(1)*                    (global src)
// arg1 assumed symmetric in LDS (addrspace(3)); args 2,3 = imm offset, imm cpol.
typedef int v4i __attribute__((vector_size(16)));
typedef __attribute__((address_space(1))) v4i g_v4i;
typedef __attribute__((address_space(3))) v4i l_v4i;
typedef __attribute__((address_space(1))) int g_i32;
typedef __attribute__((address_space(3))) int l_i32;

__device__ __forceinline__ void async_g2l_b128(const void* g, void* l) {
#if __has_builtin(__builtin_amdgcn_global_load_async_to_lds_b128)
  __builtin_amdgcn_global_load_async_to_lds_b128(
      (g_v4i*)(uintptr_t)g, (l_v4i*)(uint32_t)(uintptr_t)l, 0, 0);
#else
  asm volatile("global_load_async_to_lds_b128 %0, %1, off"
               :: "v"((uint32_t)(uintptr_t)l), "v"((uint64_t)(uintptr_t)g)
               : "memory");
#endif
}

__device__ __forceinline__ void async_g2l_b32(const void* g, void* l) {
#if __has_builtin(__builtin_amdgcn_global_load_async_to_lds_b32)
  __builtin_amdgcn_global_load_async_to_lds_b32(
      (g_i32*)(uintptr_t)g, (l_i32*)(uint32_t)(uintptr_t)l, 0, 0);
#else
  asm volatile("global_load_async_to_lds_b32 %0, %1, off"
               :: "v"((uint32_t)(uintptr_t)l), "v"((uint64_t)(uintptr_t)g)
               : "memory");
#endif
}

__device__ __forceinline__ void wait_async0() {
#if __has_builtin(__builtin_amdgcn_s_wait_asynccnt)
  __builtin_amdgcn_s_wait_asynccnt(0);
#else
  asm volatile("s_wait_asynccnt 0" ::: "memory");
#endif
}

// ---------------- parameter pool layout (dict order of setup_inputs) ----------------
#define NARR 41
#define SZ_LIST {0, 35,250,100,100,2100,30,2100,30, \
  5,5,5,5, 10,10,10,10, 10,10,10,10, 10,10,10,10,   \
  30,30,30,30, 1,1,1,1, 30,30,30,30, 1,1,1,1}

constexpr int SZ[NARR] = SZ_LIST;
__device__ const int dSZ[NARR] = SZ_LIST;

constexpr int poff(int k) { int s = 0; for (int i = 1; i < k; ++i) s += SZ[i]; return s; }

constexpr int OW1  = poff(1),  OW2  = poff(2),  OWR1 = poff(3),  OWR2 = poff(4);
constexpr int OWA1 = poff(5),  OWA2 = poff(6),  OWI1 = poff(7),  OWI2 = poff(8);
constexpr int OG1  = poff(9),  OB1  = poff(10), OM1  = poff(11), OV1  = poff(12);
constexpr int OG2  = poff(13), OB2  = poff(14), OM2  = poff(15), OV2  = poff(16);
constexpr int OGR1 = poff(17), OBR1 = poff(18), OMR1 = poff(19), OVR1 = poff(20);
constexpr int OGR2 = poff(21), OBR2 = poff(22), OMR2 = poff(23), OVR2 = poff(24);
constexpr int OGA1 = poff(25), OBA1 = poff(26), OMA1 = poff(27), OVA1 = poff(28);
constexpr int OGA2 = poff(29), OBA2 = poff(30), OMA2 = poff(31), OVA2 = poff(32);
constexpr int OGI1 = poff(33), OBI1 = poff(34), OMI1 = poff(35), OVI1 = poff(36);
constexpr int OGI2 = poff(37), OBI2 = poff(38), OMI2 = poff(39), OVI2 = poff(40);
constexpr int POOLN = poff(41);   // 5133 floats

struct TauArgs {
  const float* p[NARR];
  float* out;
  int N;
};

// ---------------- fused TauNet forward: one sample per lane ----------------
__global__ __launch_bounds__(BS) void taunet_fused(TauArgs a) {
  __shared__ float pool[POOLN];          // all weights + BN params (~20.5 KB)
  __shared__ float xs[BS * XPITCH];      // x tile, padded rows     (~21 KB)

  const int tid = threadIdx.x;
  const int gid = blockIdx.x * BS + tid;
  const int N   = a.N;

  // ---- stage all parameters into LDS via async DMA (b32 per element) ----
  {
    int off = 0;
    for (int k = 1; k < NARR; ++k) {
      const float* src = a.p[k];
      const int n = dSZ[k];
      for (int i = tid; i < n; i += BS)
        async_g2l_b32(src + i, &pool[off + i]);
      off += n;
    }
  }
  // ---- stage this block's x rows (coalesced b128 chunks, padded LDS rows) ----
  {
    const float* xg = a.p[0];
    const int brow = blockIdx.x * BS;
#pragma unroll
    for (int i = 0; i < 20; ++i) {
      const int lin = tid + i * BS;     // 16B-chunk index within 64x80 tile
      const int row = lin / 20;
      const int col = (lin % 20) * 4;   // float offset within row
      int grow = brow + row; if (grow > N - 1) grow = N - 1;   // clamp (no OOB)
      async_g2l_b128(xg + (size_t)grow * 80 + col, &xs[row * XPITCH + col]);
    }
  }
  wait_async0();      // wave's own async ops complete (data in LDS)
  __syncthreads();    // make all waves' staging visible

  const float* xrow = &xs[tid * XPITCH];

  // BN fold: y = relu(-sad * sc + sh), sh = b - m*sc
  float sc1[5], sh1[5];
#pragma unroll
  for (int c = 0; c < 5; ++c) {
    const float s = pool[OG1 + c] / sqrtf(pool[OV1 + c] + EPSV);
    sc1[c] = s; sh1[c] = pool[OB1 + c] - pool[OM1 + c] * s;
  }
  float sc2[10], sh2[10];
#pragma unroll
  for (int c = 0; c < 10; ++c) {
    const float s = pool[OG2 + c] / sqrtf(pool[OV2 + c] + EPSV);
    sc2[c] = s; sh2[c] = pool[OB2 + c] - pool[OM2 + c] * s;
  }

  // ---- layer1 (1->5, k=7, s=3, P=25) fused with layer2 (5->10, k=5, s=3, P=7) ----
  // layer2 position q consumes layer1 columns 3q..3q+4 -> x[9q..9q+18]; recompute overlap.
  float h[10][7];
#pragma unroll
  for (int q = 0; q < 7; ++q) {
    float xw[19];
#pragma unroll
    for (int i = 0; i < 19; ++i) xw[i] = xrow[9 * q + i];

    float h1l[5][5];                     // [channel][local column]
#pragma unroll
    for (int pp = 0; pp < 5; ++pp) {
#pragma unroll
      for (int c = 0; c < 5; ++c) {
        float s = 0.f;
#pragma unroll
        for (int j = 0; j < 7; ++j)
          s += fabsf(xw[3 * pp + j] - pool[OW1 + c * 7 + j]);
        h1l[c][pp] = fmaxf(-s * sc1[c] + sh1[c], 0.f);
      }
    }
#pragma unroll
    for (int c2 = 0; c2 < 10; ++c2) {
      float s = 0.f;
#pragma unroll
      for (int c = 0; c < 5; ++c)
#pragma unroll
        for (int j = 0; j < 5; ++j)
          s += fabsf(h1l[c][j] - pool[OW2 + (c2 * 5 + c) * 5 + j]);
      h[c2][q] = fmaxf(-s * sc2[c2] + sh2[c2], 0.f);
    }
  }

  // ---- residual block: r = relu(bn(sad(h,wr1))); h = relu(bn(sad(r,wr2)) + h) ----
  float r[10][7];
#pragma unroll
  for (int co = 0; co < 10; ++co) {
    const float sc = pool[OGR1 + co] / sqrtf(pool[OVR1 + co] + EPSV);
    const float sh = pool[OBR1 + co] - pool[OMR1 + co] * sc;
#pragma unroll
    for (int q = 0; q < 7; ++q) {
      float s = 0.f;
#pragma unroll
      for (int ci = 0; ci < 10; ++ci)
        s += fabsf(h[ci][q] - pool[OWR1 + co * 10 + ci]);
      r[co][q] = fmaxf(-s * sc + sh, 0.f);
    }
  }
#pragma unroll
  for (int co = 0; co < 10; ++co) {
    const float sc = pool[OGR2 + co] / sqrtf(pool[OVR2 + co] + EPSV);
    const float sh = pool[OBR2 + co] - pool[OMR2 + co] * sc;
#pragma unroll
    for (int q = 0; q < 7; ++q) {
      float s = 0.f;
#pragma unroll
      for (int ci = 0; ci < 10; ++ci)
        s += fabsf(r[ci][q] - pool[OWR2 + co * 10 + ci]);
      h[co][q] = fmaxf((-s * sc + sh) + h[co][q], 0.f);   // safe in-place: reads own index only
    }
  }

  // ---- heads: z[70] = h flattened; 70->30 SAD+BN+relu fused with 30->1 SAD+BN+relu ----
  auto head = [&](int OWx1, int OGx1, int OBx1, int OMx1, int OVx1,
                  int OWx2, int OGx2, int OBx2, int OMx2, int OVx2) -> float {
    float sacc = 0.f;
#pragma unroll 1
    for (int o = 0; o < 30; ++o) {
      float s = 0.f;
#pragma unroll
      for (int c = 0; c < 70; ++c)
        s += fabsf(h[c / 7][c % 7] - pool[OWx1 + o * 70 + c]);
      const float sc = pool[OGx1 + o] / sqrtf(pool[OVx1 + o] + EPSV);
      float av = -s * sc + (pool[OBx1 + o] - pool[OMx1 + o] * sc);
      av = fmaxf(av, 0.f);
      sacc += fabsf(av - pool[OWx2 + o]);
    }
    const float sc = pool[OGx2] / sqrtf(pool[OVx2] + EPSV);
    return fmaxf(-sacc * sc + (pool[OBx2] - pool[OMx2] * sc), 0.f);
  };

  const float ta = head(OWA1, OGA1, OBA1, OMA1, OVA1, OWA2, OGA2, OBA2, OMA2, OVA2);
  const float ti = head(OWI1, OGI1, OBI1, OMI1, OVI1, OWI2, OGI2, OBI2, OMI2, OVI2);

  if (gid < N) {
    a.out[gid]     = ta;   // tau_amp
    a.out[N + gid] = ti;   // tau_inten
  }
}

extern "C" void kernel_launch(void* const* d_in, const int* in_sizes, int n_in,
                              void* d_out, int out_size, void* d_ws, size_t ws_size,
                              hipStream_t stream) {
  (void)n_in; (void)out_size; (void)d_ws; (void)ws_size;
  TauArgs a;
  for (int i = 0; i < NARR; ++i) a.p[i] = (const float*)d_in[i];
  a.out = (float*)d_out;
  a.N   = in_sizes[0] / 80;            // x is (N, 1, 80)
  const int blocks = (a.N + BS - 1) / BS;
  taunet_fused<<<blocks, BS, 0, stream>>>(a);
}